// BPR_79431125172650
// MI455X (gfx1250) — compile-verified
//
#include <hip/hip_runtime.h>
#include <hip/hip_bf16.h>

// ---------------------------------------------------------------------------
// LightGCN 3-hop propagation for MI455X (gfx1250, wave32).
// SpMM is atomic/L2-BW bound (~0.25 FLOP/byte). CDNA5 paths used:
//   * async global->LDS row gather (ASYNCcnt pipeline)
//   * V_WMMA_F32_16X16X4_F32 performs the val_e * x[col_e] scaling
//     (A = diag(vals) over 4 edges, B = 4 gathered rows x 16 cols)
//   * native device-scope non-returning global_atomic_add_f32 scatter
// ---------------------------------------------------------------------------

typedef float v2f __attribute__((ext_vector_type(2)));
typedef float v8f __attribute__((ext_vector_type(8)));

#define DFEAT 64          // feature dim (reference D)
#define BATCH 32          // edges per wave batch
#define WPB   8           // waves per 256-thread block (wave32)
#define SPMM_BLOCKS 4096

#if defined(__has_builtin)
#if __has_builtin(__builtin_amdgcn_wmma_f32_16x16x4_f32)
#define HAVE_WMMA_F32X4 1
#endif
#endif

// ---- CDNA5 primitives (inline asm, probe-verified in round 1) -------------

__device__ __forceinline__ void async_gather_b64(unsigned lds_addr,
                                                 unsigned long long gaddr) {
  asm volatile("global_load_async_to_lds_b64 %0, %1, off"
               :: "v"(lds_addr), "v"(gaddr) : "memory");
}
__device__ __forceinline__ void wait_async_le0() {
  asm volatile("s_wait_asynccnt 0" ::: "memory");
}
__device__ __forceinline__ void atomic_add_f32_dev(float* p, float v) {
  asm volatile("global_atomic_add_f32 %0, %1, off scope:SCOPE_DEV"
               :: "v"((unsigned long long)(size_t)p), "v"(v) : "memory");
}
__device__ __forceinline__ float rdlane_f(float v, int l) {
  return __int_as_float(__builtin_amdgcn_readlane(__float_as_int(v), l));
}

// ---- SpMM: out[rows[e]] += scale * vals[e] * x[cols[e]]  (D=64, f32) ------

__global__ __launch_bounds__(256) void spmm_kernel(
    const int* __restrict__ rows, const int* __restrict__ cols,
    const float* __restrict__ vals, const float* __restrict__ x,
    float* __restrict__ out, int nE, float scale)
{
  __shared__ float lbuf[WPB][BATCH][DFEAT];    // 64KB: 8 waves * 32 rows * 256B
  const int lane  = threadIdx.x & 31;
  const int wave  = threadIdx.x >> 5;
  const int gwave = blockIdx.x * WPB + wave;
  const int nwv   = gridDim.x * WPB;
  const unsigned lds0 = (unsigned)(size_t)(&lbuf[wave][0][0]); // LDS byte addr
  const unsigned loff = (unsigned)(lane * 8);

  for (int base = gwave * BATCH; base < nE; base += nwv * BATCH) {
    const int nb  = min(BATCH, nE - base);
    const int nbp = (nb + 3) & ~3;             // pad to a 4-edge WMMA group

    int r = 0, c = 0; float v = 0.0f;          // padded lanes: row0/col0/val0
    if (lane < nb) {
      r = rows[base + lane];
      c = cols[base + lane];
      v = vals[base + lane] * scale;           // fold hop weight into val
    }

    // Async-gather the nbp rows (256B each) into LDS; one op per row.
    for (int p = 0; p < nbp; ++p) {
      const int cj = __builtin_amdgcn_readlane(c, p);
      const unsigned long long ga =
          (unsigned long long)(size_t)(x + (size_t)cj * DFEAT) + loff;
      async_gather_b64(lds0 + (unsigned)(p * DFEAT * 4) + loff, ga);
    }
    wait_async_le0();

#if defined(HAVE_WMMA_F32X4)
    const int khalf = lane >> 4;               // 0: lanes 0-15, 1: lanes 16-31
    const int ncol  = lane & 15;
    for (int g4 = 0; g4 < nbp; g4 += 4) {      // 4 edges per WMMA group
      const float s0 = rdlane_f(v, g4 + 0), s1 = rdlane_f(v, g4 + 1);
      const float s2 = rdlane_f(v, g4 + 2), s3 = rdlane_f(v, g4 + 3);
      const int   r0 = __builtin_amdgcn_readlane(r, g4 + 0);
      const int   r1 = __builtin_amdgcn_readlane(r, g4 + 1);
      const int   r2 = __builtin_amdgcn_readlane(r, g4 + 2);
      const int   r3 = __builtin_amdgcn_readlane(r, g4 + 3);

      // A(16x4): A[e][e] = val_e, rows 4-15 zero.
      v2f a;
      a.x = (lane == 0) ? s0 : (lane == 18) ? s2 : 0.0f;   // K=0 / K=2
      a.y = (lane == 1) ? s1 : (lane == 19) ? s3 : 0.0f;   // K=1 / K=3
#pragma unroll
      for (int ch = 0; ch < 4; ++ch) {         // 16-column chunks of D=64
        // B(4x16): VGPR0 = rows K=0/1, VGPR1 = rows K=2/3 across lane halves.
        v2f b;
        b.x = lbuf[wave][g4 + khalf    ][ch * 16 + ncol];
        b.y = lbuf[wave][g4 + 2 + khalf][ch * 16 + ncol];
        v8f cacc = {0.f, 0.f, 0.f, 0.f, 0.f, 0.f, 0.f, 0.f};
        v8f d = __builtin_amdgcn_wmma_f32_16x16x4_f32(
            false, a, false, b, (short)0, cacc, false, false);
        // D rows 0..3 (VGPR e, lanes 0-15) = val_e * x_row_e[ch*16 + ncol].
        if (lane < 16) {
          atomic_add_f32_dev(out + (size_t)r0 * DFEAT + ch * 16 + ncol, d[0]);
          atomic_add_f32_dev(out + (size_t)r1 * DFEAT + ch * 16 + ncol, d[1]);
          atomic_add_f32_dev(out + (size_t)r2 * DFEAT + ch * 16 + ncol, d[2]);
          atomic_add_f32_dev(out + (size_t)r3 * DFEAT + ch * 16 + ncol, d[3]);
        }
      }
    }
#else
    // Scalar fallback: readlane broadcast + fma + atomic scatter.
    for (int j = 0; j < nb; ++j) {
      const int   rj = __builtin_amdgcn_readlane(r, j);
      const float vj = rdlane_f(v, j);
      const float2 q = *(const float2*)(&lbuf[wave][j][lane * 2]);
      float* op = out + (size_t)rj * DFEAT + (unsigned)(lane * 2);
      atomic_add_f32_dev(op,     vj * q.x);
      atomic_add_f32_dev(op + 1, vj * q.y);
    }
#endif
  }
}

// ---- out = [user_emb ; item_emb], gz = 0 (float4 streaming) ---------------
__global__ void init_kernel(float4* __restrict__ out,
                            const float4* __restrict__ ue,
                            const float4* __restrict__ ie,
                            float4* __restrict__ gz, int nU4, int nTot4)
{
  const float4 z = make_float4(0.f, 0.f, 0.f, 0.f);
  for (int i = blockIdx.x * blockDim.x + threadIdx.x; i < nTot4;
       i += gridDim.x * blockDim.x) {
    out[i] = (i < nU4) ? ue[i] : ie[i - nU4];
    gz[i]  = z;
  }
}

// ---- out += w*g ; z = 0 (fused zeroing of next hop's accumulator) ---------
__global__ void axpy_zero_kernel(float4* __restrict__ out,
                                 const float4* __restrict__ g, float w,
                                 float4* __restrict__ z, int n4)
{
  const float4 zz = make_float4(0.f, 0.f, 0.f, 0.f);
  for (int i = blockIdx.x * blockDim.x + threadIdx.x; i < n4;
       i += gridDim.x * blockDim.x) {
    float4 o = out[i];
    const float4 a = g[i];
    o.x += w * a.x; o.y += w * a.y; o.z += w * a.z; o.w += w * a.w;
    out[i] = o;
    z[i]   = zz;
  }
}

// ---------------------------------------------------------------------------

extern "C" void kernel_launch(void* const* d_in, const int* in_sizes, int n_in,
                              void* d_out, int out_size, void* d_ws, size_t ws_size,
                              hipStream_t stream)
{
  const float* user_emb = (const float*)d_in[0];
  const float* item_emb = (const float*)d_in[1];
  const int*   ui_rows  = (const int*)  d_in[2];
  const int*   ui_cols  = (const int*)  d_in[3];
  const float* ui_vals  = (const float*)d_in[4];
  const float* iu_vals  = (const float*)d_in[5];

  const int U = in_sizes[0] / DFEAT;
  const int I = in_sizes[1] / DFEAT;
  const int E = in_sizes[2];

  float* out   = (float*)d_out;
  float* out_u = out;
  float* out_i = out + (size_t)U * DFEAT;

  const size_t N  = (size_t)(U + I) * DFEAT;   // elems per hop buffer
  float* g1   = (float*)d_ws;                  // [ (U+I)*D ]
  float* g2   = g1 + N;                        // [ (U+I)*D ]  (needs 153.6MB ws)
  float* g1_u = g1;  float* g1_i = g1 + (size_t)U * DFEAT;
  float* g2_u = g2;  float* g2_i = g2 + (size_t)U * DFEAT;

  const int n4  = (int)(N / 4);
  const int nU4 = (int)((size_t)U * DFEAT / 4);
  int eb = (n4 + 255) / 256;
  if (eb > 18750) eb = 18750;

  // out = emb ; g1 = 0
  init_kernel<<<eb, 256, 0, stream>>>((float4*)out, (const float4*)user_emb,
                                      (const float4*)item_emb, (float4*)g1,
                                      nU4, n4);
  // hop 1
  spmm_kernel<<<SPMM_BLOCKS, 256, 0, stream>>>(ui_rows, ui_cols, ui_vals,
                                               item_emb, g1_u, E, 1.0f);
  spmm_kernel<<<SPMM_BLOCKS, 256, 0, stream>>>(ui_cols, ui_rows, iu_vals,
                                               user_emb, g1_i, E, 1.0f);
  // out += 0.5*g1 ; g2 = 0
  axpy_zero_kernel<<<eb, 256, 0, stream>>>((float4*)out, (const float4*)g1,
                                           0.5f, (float4*)g2, n4);
  // hop 2
  spmm_kernel<<<SPMM_BLOCKS, 256, 0, stream>>>(ui_rows, ui_cols, ui_vals,
                                               g1_i, g2_u, E, 1.0f);
  spmm_kernel<<<SPMM_BLOCKS, 256, 0, stream>>>(ui_cols, ui_rows, iu_vals,
                                               g1_u, g2_i, E, 1.0f);
  // out += (1/3)*g2 ; (re-zero g1, harmless)
  axpy_zero_kernel<<<eb, 256, 0, stream>>>((float4*)out, (const float4*)g2,
                                           1.0f / 3.0f, (float4*)g1, n4);
  // hop 3: scatter 0.25-scaled contributions directly into the output
  spmm_kernel<<<SPMM_BLOCKS, 256, 0, stream>>>(ui_rows, ui_cols, ui_vals,
                                               g2_i, out_u, E, 0.25f);
  spmm_kernel<<<SPMM_BLOCKS, 256, 0, stream>>>(ui_cols, ui_rows, iu_vals,
                                               g2_u, out_i, E, 0.25f);
}